// Renderer_17935783428527
// MI455X (gfx1250) — compile-verified
//
#include <hip/hip_runtime.h>
#include <stdint.h>

#define N_RAYS 65536
#define N_SAMPLES 192
#define RAYS_PER_BLOCK 8   // 8 waves of 32 per block
#define SPL 6              // samples per lane = 192 / 32
#define DELTA_FAR 1e10f

// Force a wave-uniform 64-bit value into SGPRs.
__device__ __forceinline__ uint64_t uniform_u64(const void* p) {
    uint64_t x = (uint64_t)(uintptr_t)p;
    uint32_t lo = __builtin_amdgcn_readfirstlane((uint32_t)x);
    uint32_t hi = __builtin_amdgcn_readfirstlane((uint32_t)(x >> 32));
    return ((uint64_t)hi << 32) | (uint64_t)lo;
}

// CDNA5 async data mover (GVS form): saddr base + 32-bit lane offset.
// Per CDNA5 ISA, the instruction immediate offset is added to BOTH
// the LDS destination and the global source address -> contiguous->contiguous
// chunk copies need only one immediate per issue.
template <int IMM>
__device__ __forceinline__ void async_b128(uint32_t lds, uint32_t voff, uint64_t sbase) {
    asm volatile("global_load_async_to_lds_b128 %0, %1, %2 offset:%3 th:TH_LOAD_NT"
                 :: "v"(lds), "v"(voff), "s"(sbase), "i"(IMM) : "memory");
}
template <int IMM>
__device__ __forceinline__ void async_b64(uint32_t lds, uint32_t voff, uint64_t sbase) {
    asm volatile("global_load_async_to_lds_b64 %0, %1, %2 offset:%3 th:TH_LOAD_NT"
                 :: "v"(lds), "v"(voff), "s"(sbase), "i"(IMM) : "memory");
}

__device__ __forceinline__ void wait_async0() {
    asm volatile("s_wait_asynccnt 0" ::: "memory");
}

__global__ __launch_bounds__(256) void nerf_render_kernel(
    const float* __restrict__ sigma,   // (R, S, 1)
    const float* __restrict__ color,   // (R, S, 3)
    const float* __restrict__ t,       // (R, S, 1)
    float* __restrict__ out)           // rgb (R*3) then depth (R)
{
    __shared__ alignas(16) float s_sig[RAYS_PER_BLOCK][N_SAMPLES];
    __shared__ alignas(16) float s_t  [RAYS_PER_BLOCK][N_SAMPLES];
    __shared__ alignas(16) float s_col[RAYS_PER_BLOCK][N_SAMPLES * 3];

    const int wv   = threadIdx.x >> 5;
    const int lane = threadIdx.x & 31;
    const int ray  = blockIdx.x * RAYS_PER_BLOCK + wv;

    // Wave-uniform global bases in SGPRs (ray is uniform within a wave).
    const uint64_t gsig = uniform_u64(sigma + (size_t)ray * N_SAMPLES);
    const uint64_t gt   = uniform_u64(t     + (size_t)ray * N_SAMPLES);
    const uint64_t gcol = uniform_u64(color + (size_t)ray * N_SAMPLES * 3);

    // Per-lane LDS destinations / global offsets.
    const uint32_t off128 = (uint32_t)(lane * 16);
    const uint32_t off64  = (uint32_t)(lane * 8);
    const uint32_t lsig   = (uint32_t)(uintptr_t)&s_sig[wv][0];
    const uint32_t lt     = (uint32_t)(uintptr_t)&s_t  [wv][0];
    const uint32_t lcol   = (uint32_t)(uintptr_t)&s_col[wv][0];

    // ---- async stage, full-EXEC wave-wide chunks (9 issues / ray) ----
    // sigma: 768 B = 512 (b128) + 256 (b64)
    async_b128<0>  (lsig + off128, off128, gsig);
    async_b64 <512>(lsig + off64,  off64,  gsig);
    // t: 768 B
    async_b128<0>  (lt + off128, off128, gt);
    async_b64 <512>(lt + off64,  off64,  gt);
    // color: 2304 B = 4*512 (b128) + 256 (b64)
    async_b128<0>   (lcol + off128, off128, gcol);
    async_b128<512> (lcol + off128, off128, gcol);
    async_b128<1024>(lcol + off128, off128, gcol);
    async_b128<1536>(lcol + off128, off128, gcol);
    async_b64 <2048>(lcol + off64,  off64,  gcol);

    wait_async0();  // this wave's LDS data is ready; no cross-wave sharing

    // ---- per-lane 6 contiguous samples ----
    const int s0 = lane * SPL;
    float tt[SPL], om[SPL], al[SPL];
    #pragma unroll
    for (int j = 0; j < SPL; ++j) tt[j] = s_t[wv][s0 + j];

    const float t_next = (lane < 31) ? s_t[wv][s0 + SPL] : 0.0f;

    #pragma unroll
    for (int j = 0; j < SPL; ++j) {
        float sg = s_sig[wv][s0 + j];
        sg = sg > 0.0f ? sg : 0.0f;                       // relu
        float delta;
        if (j < SPL - 1)      delta = tt[j + 1] - tt[j];
        else if (lane < 31)   delta = t_next - tt[j];
        else                  delta = DELTA_FAR;          // last sample
        const float o = __expf(-sg * delta);              // = 1 - alpha
        om[j] = o;
        al[j] = 1.0f - o;
    }

    // ---- exclusive cumprod of (1 - alpha) across the wave ----
    float x = om[0] * om[1] * om[2] * om[3] * om[4] * om[5];  // lane-local product
    #pragma unroll
    for (int off = 1; off < 32; off <<= 1) {
        const float y = __shfl_up(x, off, 32);
        if (lane >= off) x *= y;
    }
    float T = __shfl_up(x, 1, 32);    // exclusive prefix product
    if (lane == 0) T = 1.0f;

    // ---- weighted accumulation ----
    float r = 0.f, g = 0.f, b = 0.f, d = 0.f;
    #pragma unroll
    for (int j = 0; j < SPL; ++j) {
        const int s = s0 + j;
        const float wgt = al[j] * T;
        r += wgt * s_col[wv][3 * s + 0];
        g += wgt * s_col[wv][3 * s + 1];
        b += wgt * s_col[wv][3 * s + 2];
        d += wgt * tt[j];
        T *= om[j];
    }

    // ---- wave32 reduction ----
    #pragma unroll
    for (int off = 16; off > 0; off >>= 1) {
        r += __shfl_xor(r, off, 32);
        g += __shfl_xor(g, off, 32);
        b += __shfl_xor(b, off, 32);
        d += __shfl_xor(d, off, 32);
    }

    if (lane == 0) {
        out[(size_t)ray * 3 + 0] = r;
        out[(size_t)ray * 3 + 1] = g;
        out[(size_t)ray * 3 + 2] = b;
        out[(size_t)N_RAYS * 3 + ray] = d;
    }
}

extern "C" void kernel_launch(void* const* d_in, const int* in_sizes, int n_in,
                              void* d_out, int out_size, void* d_ws, size_t ws_size,
                              hipStream_t stream) {
    const float* sigma = (const float*)d_in[0];
    const float* color = (const float*)d_in[1];
    const float* t     = (const float*)d_in[2];
    float* out = (float*)d_out;

    dim3 grid(N_RAYS / RAYS_PER_BLOCK);
    dim3 block(256);
    hipLaunchKernelGGL(nerf_render_kernel, grid, block, 0, stream,
                       sigma, color, t, out);
}